// DeformableConv2dLayer_57131654971629
// MI455X (gfx1250) — compile-verified
//
#include <hip/hip_runtime.h>
#include <math.h>

// ---------------- problem constants (match reference) ----------------
#define K2      9
#define CIN     32
#define COUT    32
#define HH      192
#define WW      192
#define BB      4
#define KTOT    (CIN * K2)     // 288 = GEMM reduction depth (k = t*32 + c permuted order)
#define LDSTR   292            // padded LDS row stride (floats) -> conflict-free frag reads
#define TILESX  (WW / 16)      // 12 sixteen-pixel segments per row
#define WAVES   4              // waves (tiles) per block

typedef float v2f __attribute__((ext_vector_type(2)));
typedef float v8f __attribute__((ext_vector_type(8)));

__device__ __forceinline__ float fast_sigmoid2(float v) {
    // 2*sigmoid(v) with v_rcp_f32 instead of the IEEE divide sequence
    return 2.0f * __builtin_amdgcn_rcpf(1.0f + __expf(-v));
}

// =====================================================================
// Kernel T: NCHW -> NHWC transpose of x.
// Reads coalesced per-channel, writes 8 x float4 (128B) per thread.
// x stays L2-resident (18.9 MB << 192 MB), so both conv kernels then
// gather whole 16-channel groups with b128 loads.
// =====================================================================
__global__ __launch_bounds__(256)
void nchw_to_nhwc_kernel(const float* __restrict__ x, float* __restrict__ xh)
{
    const int pix = blockIdx.x * 256 + threadIdx.x;    // b*H*W total, exact grid
    const int b   = pix / (HH * WW);
    const int rem = pix - b * (HH * WW);

    float v[CIN];
#pragma unroll
    for (int c = 0; c < CIN; ++c)
        v[c] = x[((size_t)b * CIN + c) * HH * WW + rem];

    float4* dst = (float4*)(xh + (size_t)pix * CIN);
#pragma unroll
    for (int i = 0; i < CIN / 4; ++i)
        dst[i] = ((const float4*)v)[i];
}

// =====================================================================
// Shared tile geometry for both WMMA kernels:
//   one wave  = one 16-pixel row segment (b, yh, x0..x0+15)
//   GEMM      : D[32 x 16] = W[32 x 288] * V[288 x 16], 72 x WMMA_F32_16X16X4_F32
//   K order   : k = t*32 + c  (so per-tap channel groups are contiguous in LDS)
// Fragment layouts (per ISA):
//   A 16x4 f32 : lane<16 -> row=lane,    {K0,K1}; lane>=16 -> row=lane-16, {K2,K3}
//   B 4x16 f32 : lane<16 -> col=lane,    {K0,K1}; lane>=16 -> col=lane-16, {K2,K3}
//   C/D 16x16  : vgpr r, lane<16 -> (M=r, N=lane); lane>=16 -> (M=r+8, N=lane-16)
// =====================================================================

// =====================================================================
// Kernel A: offsets (18 ch) + modulator (9 ch) 3x3 conv as WMMA GEMM.
// Weight matrix padded to 32 rows (rows 27..31 = 0); biases folded into
// the accumulator init (channel depends only on vgpr index + lane half).
// =====================================================================
__global__ __launch_bounds__(128)
void dcn_offset_mod_kernel(const float* __restrict__ xh,
                           const float* __restrict__ w_off,
                           const float* __restrict__ b_off,
                           const float* __restrict__ w_mod,
                           const float* __restrict__ b_mod,
                           float* __restrict__ off_out,
                           float* __restrict__ mod_out)
{
    extern __shared__ float smem[];
    float* Wlds = smem;                       // 32 * LDSTR
    float* Vlds = smem + COUT * LDSTR;        // WAVES * 16 * LDSTR

    const int tid  = threadIdx.x;
    const int wave = tid >> 5;
    const int lane = tid & 31;
    const int p    = lane & 15;
    const int half = lane >> 4;

    // Stage padded weight matrix in permuted K order:
    //   Wlds[o][t*32 + c] = w[o][c][t];  rows 0..17 = w_off, 18..26 = w_mod, 27..31 = 0
    for (int i = tid; i < 32 * KTOT; i += 128) {
        const int o = i / KTOT, r = i - o * KTOT;
        const int t = r >> 5, c = r & 31;
        float wv = 0.0f;
        if (o < 18)      wv = w_off[o * KTOT + c * 9 + t];
        else if (o < 27) wv = w_mod[(o - 18) * KTOT + c * 9 + t];
        Wlds[o * LDSTR + r] = wv;
    }

    const int tile = blockIdx.x * WAVES + wave;
    const int xt = tile % TILESX;
    const int yh = (tile / TILESX) % HH;
    const int b  = tile / (TILESX * HH);
    const int xw = xt * 16 + p;

    // ---- Phase 1: plain im2col V[p][t*32+c] from NHWC; b128 in, b128 out ----
    float* Vw = Vlds + (wave * 16 + p) * LDSTR;
    const int cbase = half * 16;

#pragma unroll
    for (int t = 0; t < 9; ++t) {
        const int iy = yh - 1 + t / 3;
        const int ix = xw - 1 + t % 3;
        const bool valid = (iy >= 0) && (iy < HH) && (ix >= 0) && (ix < WW);

        float4 q[4] = {};
        if (valid) {
            const float* src = xh + ((size_t)(b * HH + iy) * WW + ix) * CIN + cbase;
#pragma unroll
            for (int i = 0; i < 4; ++i) q[i] = *(const float4*)(src + 4 * i);
        }
        float4* dst = (float4*)(Vw + t * 32 + cbase);
#pragma unroll
        for (int i = 0; i < 4; ++i) dst[i] = q[i];
    }

    // ---- Bias-initialized accumulators (C layout is pixel-invariant) ----
    const int mbase = half * 8;
    v8f acc0, acc1;
#pragma unroll
    for (int r = 0; r < 8; ++r) {
        const int ch0 = r + mbase;            // 0..15
        const int ch1 = ch0 + 16;             // 16..31
        acc0[r] = (ch0 < 18) ? b_off[ch0] : b_mod[ch0 - 18];
        acc1[r] = (ch1 < 18) ? b_off[ch1] : ((ch1 < 27) ? b_mod[ch1 - 18] : 0.0f);
    }
    __syncthreads();

    // ---- Phase 2: K-loop of fp32 WMMA ----
    const float* Arow0 = Wlds + (lane & 15) * LDSTR;
    const float* Arow1 = Wlds + ((lane & 15) + 16) * LDSTR;
    const float* Brow  = Vlds + (wave * 16 + p) * LDSTR;
    const int koff = half * 2;

#pragma unroll 4
    for (int k = 0; k < KTOT; k += 4) {
        const v2f a0 = *(const v2f*)(Arow0 + k + koff);
        const v2f a1 = *(const v2f*)(Arow1 + k + koff);
        const v2f bf = *(const v2f*)(Brow  + k + koff);
        acc0 = __builtin_amdgcn_wmma_f32_16x16x4_f32(
                   false, a0, false, bf, (short)0, acc0, false, false);
        acc1 = __builtin_amdgcn_wmma_f32_16x16x4_f32(
                   false, a1, false, bf, (short)0, acc1, false, false);
    }

    // ---- Epilogue: rows 0..17 -> offsets; 18..26 -> 2*sigmoid -> modulator ----
#pragma unroll
    for (int r = 0; r < 8; ++r) {
        const int ch0 = r + mbase;
        if (ch0 < 18)
            off_out[((b * 18 + ch0) * HH + yh) * WW + xw] = acc0[r];
        else
            mod_out[((b * 9 + (ch0 - 18)) * HH + yh) * WW + xw] = fast_sigmoid2(acc0[r]);
        const int ch1 = ch0 + 16;
        if (ch1 < 18)
            off_out[((b * 18 + ch1) * HH + yh) * WW + xw] = acc1[r];
        else if (ch1 < 27)
            mod_out[((b * 9 + (ch1 - 18)) * HH + yh) * WW + xw] = fast_sigmoid2(acc1[r]);
    }
}

// =====================================================================
// Kernel B: deformable sampling (bilinear + mask) from NHWC into LDS,
// then D[32 x 16] = W[32 x 288] * V[288 x 16] via fp32 WMMA.
// =====================================================================
__global__ __launch_bounds__(128)
void dcn_main_kernel(const float* __restrict__ xh,
                     const float* __restrict__ w_reg,
                     const float* __restrict__ off,
                     const float* __restrict__ mod,
                     float* __restrict__ out)
{
    extern __shared__ float smem[];
    float* Wlds = smem;                       // COUT * LDSTR
    float* Vlds = smem + COUT * LDSTR;        // WAVES * 16 * LDSTR

    const int tid  = threadIdx.x;
    const int wave = tid >> 5;
    const int lane = tid & 31;
    const int p    = lane & 15;
    const int half = lane >> 4;

    // Stage weight matrix in permuted K order: Wlds[o][t*32+c] = w_reg[o][c][t]
    for (int i = tid; i < COUT * KTOT; i += 128) {
        const int o = i / KTOT, r = i - o * KTOT;
        const int t = r >> 5, c = r & 31;
        Wlds[o * LDSTR + r] = w_reg[o * KTOT + c * 9 + t];
    }

    const int tile = blockIdx.x * WAVES + wave;
    const int xt = tile % TILESX;
    const int yh = (tile / TILESX) % HH;
    const int b  = tile / (TILESX * HH);
    const int xw = xt * 16 + p;

    const float* offb = off + (size_t)b * 18 * HH * WW;
    const float* modb = mod + (size_t)b *  9 * HH * WW;
    const float* xhb  = xh  + (size_t)b * HH * WW * CIN;

    // Prefetch the per-pixel offset/modulator rows (global_prefetch_b8)
    __builtin_prefetch(offb + yh * WW + xw, 0, 1);
    __builtin_prefetch(modb + yh * WW + xw, 0, 1);

    // ---- Phase 1: deformed, modulated V[p][t*32+c]; 16 b128 gathers/tap ----
    float* Vw = Vlds + (wave * 16 + p) * LDSTR;
    const int cbase = half * 16;

#pragma unroll
    for (int t = 0; t < 9; ++t) {
        const float oy = offb[((2 * t)     * HH + yh) * WW + xw];
        const float ox = offb[((2 * t + 1) * HH + yh) * WW + xw];
        const float m  = modb[( t          * HH + yh) * WW + xw];

        const float py = (float)(yh - 1 + t / 3) + oy;
        const float px = (float)(xw - 1 + t % 3) + ox;
        const float fy = floorf(py), fx = floorf(px);
        const float wy = py - fy,    wx = px - fx;
        const int y0 = (int)fy, x0 = (int)fx;
        const int y1 = y0 + 1,  x1 = x0 + 1;

        const bool vy0 = (y0 >= 0) && (y0 < HH), vy1 = (y1 >= 0) && (y1 < HH);
        const bool vx0 = (x0 >= 0) && (x0 < WW), vx1 = (x1 >= 0) && (x1 < WW);
        const int cy0 = min(max(y0, 0), HH - 1), cy1 = min(max(y1, 0), HH - 1);
        const int cx0 = min(max(x0, 0), WW - 1), cx1 = min(max(x1, 0), WW - 1);

        const float w00 = (vy0 && vx0) ? (1.0f - wy) * (1.0f - wx) * m : 0.0f;
        const float w01 = (vy0 && vx1) ? (1.0f - wy) * wx          * m : 0.0f;
        const float w10 = (vy1 && vx0) ? wy          * (1.0f - wx) * m : 0.0f;
        const float w11 = (vy1 && vx1) ? wy * wx                   * m : 0.0f;

        const float* s00 = xhb + ((size_t)cy0 * WW + cx0) * CIN + cbase;
        const float* s01 = xhb + ((size_t)cy0 * WW + cx1) * CIN + cbase;
        const float* s10 = xhb + ((size_t)cy1 * WW + cx0) * CIN + cbase;
        const float* s11 = xhb + ((size_t)cy1 * WW + cx1) * CIN + cbase;

        float4* dst = (float4*)(Vw + t * 32 + cbase);
#pragma unroll
        for (int i = 0; i < 4; ++i) {
            const float4 q00 = *(const float4*)(s00 + 4 * i);
            const float4 q01 = *(const float4*)(s01 + 4 * i);
            const float4 q10 = *(const float4*)(s10 + 4 * i);
            const float4 q11 = *(const float4*)(s11 + 4 * i);
            float4 vr;
            vr.x = w00 * q00.x + w01 * q01.x + w10 * q10.x + w11 * q11.x;
            vr.y = w00 * q00.y + w01 * q01.y + w10 * q10.y + w11 * q11.y;
            vr.z = w00 * q00.z + w01 * q01.z + w10 * q10.z + w11 * q11.z;
            vr.w = w00 * q00.w + w01 * q01.w + w10 * q10.w + w11 * q11.w;
            dst[i] = vr;                       // ds_store_b128
        }
    }
    __syncthreads();

    // ---- Phase 2: K-loop of fp32 WMMA ----
    const float* Arow0 = Wlds + (lane & 15) * LDSTR;
    const float* Arow1 = Wlds + ((lane & 15) + 16) * LDSTR;
    const float* Brow  = Vlds + (wave * 16 + p) * LDSTR;
    const int koff = half * 2;

    v8f acc0 = {};
    v8f acc1 = {};
#pragma unroll 4
    for (int k = 0; k < KTOT; k += 4) {
        const v2f a0 = *(const v2f*)(Arow0 + k + koff);
        const v2f a1 = *(const v2f*)(Arow1 + k + koff);
        const v2f bf = *(const v2f*)(Brow  + k + koff);
        acc0 = __builtin_amdgcn_wmma_f32_16x16x4_f32(
                   false, a0, false, bf, (short)0, acc0, false, false);
        acc1 = __builtin_amdgcn_wmma_f32_16x16x4_f32(
                   false, a1, false, bf, (short)0, acc1, false, false);
    }

    // ---- Phase 3: scatter D per the 16x16 f32 C/D layout ----
    const int mbase = half * 8;
#pragma unroll
    for (int r = 0; r < 8; ++r) {
        const int ch = r + mbase;
        out[(((size_t)b * COUT + ch     ) * HH + yh) * WW + xw] = acc0[r];
        out[(((size_t)b * COUT + ch + 16) * HH + yh) * WW + xw] = acc1[r];
    }
}

// =====================================================================
// Launch
// =====================================================================
extern "C" void kernel_launch(void* const* d_in, const int* in_sizes, int n_in,
                              void* d_out, int out_size, void* d_ws, size_t ws_size,
                              hipStream_t stream) {
    (void)in_sizes; (void)n_in; (void)out_size; (void)ws_size;

    const float* x     = (const float*)d_in[0];
    const float* w_off = (const float*)d_in[1];
    const float* b_off = (const float*)d_in[2];
    const float* w_mod = (const float*)d_in[3];
    const float* b_mod = (const float*)d_in[4];
    const float* w_reg = (const float*)d_in[5];
    float* out = (float*)d_out;

    // workspace: NHWC x, then offsets, then modulator (~35 MB total)
    float* xh_buf  = (float*)d_ws;                                   // B*H*W*32
    float* off_buf = xh_buf  + (size_t)BB * HH * WW * CIN;           // B*18*H*W
    float* mod_buf = off_buf + (size_t)BB * 18 * HH * WW;            // B* 9*H*W

    const int nPix   = BB * HH * WW;              // 147456
    const int nTiles = BB * HH * TILESX;          // 9216
    const size_t shmem = (size_t)(COUT * LDSTR + WAVES * 16 * LDSTR) * sizeof(float);

    // T: NCHW -> NHWC
    nchw_to_nhwc_kernel<<<nPix / 256, 256, 0, stream>>>(x, xh_buf);

    // A: offset/modulator conv on the matrix pipe
    dcn_offset_mod_kernel<<<nTiles / WAVES, 128, shmem, stream>>>(
        xh_buf, w_off, b_off, w_mod, b_mod, off_buf, mod_buf);

    // B: deformable gather + main GEMM on the matrix pipe
    dcn_main_kernel<<<nTiles / WAVES, 128, shmem, stream>>>(
        xh_buf, w_reg, off_buf, mod_buf, out);
}